// Dist_Loss_39118562132223
// MI455X (gfx1250) — compile-verified
//
#include <hip/hip_runtime.h>
#include <hip/hip_bf16.h>

typedef __attribute__((ext_vector_type(2))) float v2f;
typedef __attribute__((ext_vector_type(8))) float v8f;

#define NPTS 4096
#define DIM  64
#define TILES_PER_DIM (NPTS / 16)      // 256
#define WAVES_PER_BLOCK 8

// workspace layout (float indices)
#define WS_SUMS      0      // [0]=same_src [1]=diff_src [2]=same_tgt [3]=diff_tgt
#define WS_CSUM_SRC  4      // 4 classes * 64 dims
#define WS_CSUM_TGT  260
#define WS_CNT_SRC   516
#define WS_CNT_TGT   520
#define WS_ZERO_N    524
#define WS_NORM_SRC  1024   // 4096 row norms
#define WS_NORM_TGT  5120

__global__ void zero_ws_kernel(float* __restrict__ ws) {
    int i = blockIdx.x * blockDim.x + threadIdx.x;
    if (i < WS_ZERO_N) ws[i] = 0.0f;
}

__global__ void row_norms_kernel(const float* __restrict__ X, float* __restrict__ norms) {
    int row = blockIdx.x * blockDim.x + threadIdx.x;
    if (row >= NPTS) return;
    const float4* p4 = (const float4*)(X + row * DIM);
    float s = 0.0f;
#pragma unroll
    for (int d = 0; d < DIM / 4; ++d) {
        float4 v = p4[d];
        s = fmaf(v.x, v.x, s);
        s = fmaf(v.y, v.y, s);
        s = fmaf(v.z, v.z, s);
        s = fmaf(v.w, v.w, s);
    }
    norms[row] = s;
}

__global__ void center_accum_kernel(const float* __restrict__ X, const int* __restrict__ L,
                                    float* __restrict__ csum, float* __restrict__ cnt) {
    int row = blockIdx.x * blockDim.x + threadIdx.x;
    if (row >= NPTS) return;
    int lab = L[row] & 3;
    atomicAdd(&cnt[lab], 1.0f);
    const float* p = X + row * DIM;
#pragma unroll 4
    for (int d = 0; d < DIM; ++d) atomicAdd(&csum[lab * DIM + d], p[d]);
}

// One wave computes one 16x16 tile of the Gram matrix via 16 chained
// V_WMMA_F32_16X16X4_F32 ops (K=64), then fuses the distance epilogue
// (d2 = |x|^2 + |y|^2 - 2*g, clamp, sqrt, label mask) entirely in the
// accumulator registers and reduces to two scalars per wave.
__global__ __launch_bounds__(256)
void dist_tiles_wmma_kernel(const float* __restrict__ X, const int* __restrict__ L,
                            const float* __restrict__ norms, float* __restrict__ sums) {
    const int lane = threadIdx.x & 31;           // wave32
    const int wave = threadIdx.x >> 5;
    const int tile = blockIdx.x * WAVES_PER_BLOCK + wave;
    const int m0 = (tile / TILES_PER_DIM) * 16;
    const int n0 = (tile % TILES_PER_DIM) * 16;
    const int hi = lane >> 4;                    // 0: K={k,k+1}, 1: K={k+2,k+3}
    const int lm = lane & 15;

    const float* arow = X + (m0 + lm) * DIM + 2 * hi;   // A: M = lm
    const float* brow = X + (n0 + lm) * DIM + 2 * hi;   // B = Y^T: N = lm

    v8f c = {};
#pragma unroll
    for (int k = 0; k < DIM; k += 4) {
        v2f a = *(const v2f*)(arow + k);
        v2f b = *(const v2f*)(brow + k);
        // (neg_a, A, neg_b, B, c_mod, C, reuse_a, reuse_b)
        c = __builtin_amdgcn_wmma_f32_16x16x4_f32(false, a, false, b,
                                                  (short)0, c, false, false);
    }

    // C/D layout: lane%16 = N; VGPR r holds M = r + 8*(lane>=16)
    const int n  = n0 + lm;
    const float y2 = norms[n];
    const int   ln = L[n];
    float sSame = 0.0f, sDiff = 0.0f;
#pragma unroll
    for (int r = 0; r < 8; ++r) {
        int m = m0 + r + 8 * hi;
        float d2 = norms[m] + y2 - 2.0f * c[r];
        float d  = (d2 > 0.0f) ? sqrtf(d2) : 0.0f;
        if (L[m] == ln) sSame += d; else sDiff += d;
    }

    // wave32 reduction
#pragma unroll
    for (int off = 16; off > 0; off >>= 1) {
        sSame += __shfl_down(sSame, off, 32);
        sDiff += __shfl_down(sDiff, off, 32);
    }
    if (lane == 0) {
        atomicAdd(&sums[0], sSame);
        atomicAdd(&sums[1], sDiff);
    }
}

__global__ void finalize_kernel(const float* __restrict__ ws,
                                const float* __restrict__ alpha_p,
                                float* __restrict__ out) {
    if (threadIdx.x != 0 || blockIdx.x != 0) return;
    float alpha = alpha_p[0];
    float t = (ws[WS_SUMS + 0] * 0.5f - alpha * ws[WS_SUMS + 1]) / (float)NPTS
            + (ws[WS_SUMS + 2] * 0.5f - alpha * ws[WS_SUMS + 3]) / (float)NPTS;
    float d31 = 0.0f;
    for (int cc = 0; cc < 4; ++cc) {
        float cs = ws[WS_CNT_SRC + cc];
        float ct = ws[WS_CNT_TGT + cc];
        float d2 = 0.0f;
        for (int d = 0; d < DIM; ++d) {
            float a = ws[WS_CSUM_SRC + cc * DIM + d] / cs;
            float b = ws[WS_CSUM_TGT + cc * DIM + d] / ct;
            float df = a - b;
            d2 = fmaf(df, df, d2);
        }
        d31 += (d2 > 0.0f) ? sqrtf(d2) : 0.0f;
    }
    out[0] = t + d31 * 0.25f;   // mean over 4 diagonal center distances
}

extern "C" void kernel_launch(void* const* d_in, const int* in_sizes, int n_in,
                              void* d_out, int out_size, void* d_ws, size_t ws_size,
                              hipStream_t stream) {
    const float* src  = (const float*)d_in[0];
    const int*   srcL = (const int*)  d_in[1];
    const float* tgt  = (const float*)d_in[2];
    const int*   tgtL = (const int*)  d_in[3];
    const float* alph = (const float*)d_in[4];
    float* ws  = (float*)d_ws;
    float* out = (float*)d_out;

    zero_ws_kernel<<<(WS_ZERO_N + 255) / 256, 256, 0, stream>>>(ws);

    row_norms_kernel<<<NPTS / 256, 256, 0, stream>>>(src, ws + WS_NORM_SRC);
    row_norms_kernel<<<NPTS / 256, 256, 0, stream>>>(tgt, ws + WS_NORM_TGT);

    center_accum_kernel<<<NPTS / 256, 256, 0, stream>>>(src, srcL,
                                                        ws + WS_CSUM_SRC, ws + WS_CNT_SRC);
    center_accum_kernel<<<NPTS / 256, 256, 0, stream>>>(tgt, tgtL,
                                                        ws + WS_CSUM_TGT, ws + WS_CNT_TGT);

    const int tile_blocks = (TILES_PER_DIM * TILES_PER_DIM) / WAVES_PER_BLOCK; // 8192
    dist_tiles_wmma_kernel<<<tile_blocks, 256, 0, stream>>>(src, srcL,
                                                            ws + WS_NORM_SRC, ws + WS_SUMS + 0);
    dist_tiles_wmma_kernel<<<tile_blocks, 256, 0, stream>>>(tgt, tgtL,
                                                            ws + WS_NORM_TGT, ws + WS_SUMS + 2);

    finalize_kernel<<<1, 64, 0, stream>>>(ws, alph, out);
}